// DendriticLayer_65352222376342
// MI455X (gfx1250) — compile-verified
//
#include <hip/hip_runtime.h>

typedef __attribute__((ext_vector_type(2))) float v2f;
typedef __attribute__((ext_vector_type(8))) float v8f;

#define N_IN      1024
#define N_OUT     256
#define BATCH     512
#define T_STEPS   100
#define NPAD      112                       // 7 column tiles of 16
#define TAU_SYN_INV 0.2f                    // 1/tau_syn
#define DECAY     0.81873075307798182f      // exp(-dt/tau_syn)
#define A_COEF    0.1f                      // dt/tau_mem
#define LOG_1MA  (-0.10536051565782630f)    // log(1 - dt/tau_mem)
#define INV_TAU_SOFT 2.0f                   // 1/tau_soft

// ---- DPP16 butterfly helpers (wave32; each 16-lane row reduced independently) ----
#define DPP_QUAD_XOR1  0xB1   // quad_perm [1,0,3,2]
#define DPP_QUAD_XOR2  0x4E   // quad_perm [2,3,0,1]
#define DPP_ROW_HMIRR  0x141  // row_half_mirror (combines the two quads of each 8)
#define DPP_ROW_MIRROR 0x140  // row_mirror      (combines the two halves of the 16-row)

template <int CTRL>
__device__ __forceinline__ float dpp_xchg(float x) {
    int r = __builtin_amdgcn_update_dpp(0, __builtin_bit_cast(int, x),
                                        CTRL, 0xF, 0xF, true);
    return __builtin_bit_cast(float, r);
}

__device__ __forceinline__ float row16_max(float x) {
    x = fmaxf(x, dpp_xchg<DPP_QUAD_XOR1>(x));
    x = fmaxf(x, dpp_xchg<DPP_QUAD_XOR2>(x));
    x = fmaxf(x, dpp_xchg<DPP_ROW_HMIRR>(x));
    x = fmaxf(x, dpp_xchg<DPP_ROW_MIRROR>(x));
    return x;
}

__device__ __forceinline__ float row16_sum(float x) {
    x += dpp_xchg<DPP_QUAD_XOR1>(x);
    x += dpp_xchg<DPP_QUAD_XOR2>(x);
    x += dpp_xchg<DPP_ROW_HMIRR>(x);
    x += dpp_xchg<DPP_ROW_MIRROR>(x);
    return x;
}

__global__ __launch_bounds__(256)
void dendritic_wmma_kernel(const float* __restrict__ spk,   // [B, N_IN]
                           const float* __restrict__ W,     // [N_OUT, N_IN]
                           float* __restrict__ out)         // [B, N_OUT]
{
    // B-matrix of the causal leaky-integrator filter:
    // sLt[k][n] = a*(1-a)^(n-k) for k <= n < 100, else 0.   (44.8 KB LDS)
    __shared__ float sLt[T_STEPS * NPAD];
    for (int i = threadIdx.x; i < T_STEPS * NPAD; i += 256) {
        int k = i / NPAD;
        int n = i - k * NPAD;
        float v = 0.0f;
        if (n < T_STEPS && n >= k)
            v = A_COEF * __expf((float)(n - k) * LOG_1MA);
        sLt[i] = v;
    }
    __syncthreads();

    const int lane = threadIdx.x & 31;
    const int wid  = threadIdx.x >> 5;
    const int m0   = (blockIdx.x * 8 + wid) * 16;   // 16 output rows per wave
    const int colL = lane & 15;
    const bool hi  = lane >= 16;

    // Each lane owns the PSP recurrence of row r = m0 + (lane%16);
    // both wave halves duplicate it (they supply different K slots of the A frag).
    const int r = m0 + colL;
    const int b = r >> 8;          // / N_OUT
    const int o = r & 255;         // % N_OUT
    const float4 sv = *(const float4*)(spk + b * N_IN + 4 * o);
    const float4 wv = *(const float4*)(W   + o * N_IN + 4 * o);
    float sarr[4] = {sv.x, sv.y, sv.z, sv.w};
    float wgt[4]  = {wv.x, wv.y, wv.z, wv.w};
    int   t0[4];
    float e0[4];
    float psp[4] = {0.f, 0.f, 0.f, 0.f};
    #pragma unroll
    for (int k = 0; k < 4; ++k) {
        float c = ceilf(sarr[k]);                 // first grid time with t >= s
        t0[k] = (int)c;
        e0[k] = __expf((sarr[k] - c) * TAU_SYN_INV);  // psp value at t0
    }

    v8f acc[7] = {};   // V tiles: 16 rows x 112 timesteps, f32

    for (int kb = 0; kb < T_STEPS; kb += 4) {
        // Advance PSP state 4 timesteps; I[s] = branch current at t = kb+s
        float I[4];
        #pragma unroll
        for (int s4 = 0; s4 < 4; ++s4) {
            int t = kb + s4;
            float Iv = 0.f;
            #pragma unroll
            for (int k = 0; k < 4; ++k) {
                psp[k] = (t == t0[k]) ? e0[k] : psp[k] * DECAY;
                Iv = fmaf(wgt[k], psp[k], Iv);
            }
            I[s4] = Iv;
        }
        // A fragment (16x4 f32): lanes 0-15 -> K=kb,kb+1 ; lanes 16-31 -> K=kb+2,kb+3
        v2f afrag;
        afrag.x = hi ? I[2] : I[0];
        afrag.y = hi ? I[3] : I[1];

        // B fragments (4x16 f32) from LDS: v0 row K=kb(+2), v1 row K=kb+1(+3)
        const int krow = kb + (hi ? 2 : 0);
        const float* brow = sLt + krow * NPAD + colL;
        #pragma unroll
        for (int tile = 0; tile < 7; ++tile) {
            v2f bfrag;
            bfrag.x = brow[tile * 16];
            bfrag.y = brow[NPAD + tile * 16];
            acc[tile] = __builtin_amdgcn_wmma_f32_16x16x4_f32(
                false, afrag, false, bfrag, (short)0, acc[tile], false, false);
        }
    }

    // Soft-argmax over time, fully in-register (DPP16 butterflies, no LDS).
    // C/D layout: VGPR j -> row j (lanes 0-15) / row j+8 (lanes 16-31), col = lane%16.
    #pragma unroll
    for (int j = 0; j < 8; ++j) {
        float y[7];
        float mx = -1e30f;
        #pragma unroll
        for (int tile = 0; tile < 7; ++tile) {
            y[tile] = acc[tile][j] * INV_TAU_SOFT;
            int t = tile * 16 + colL;
            if (t < T_STEPS) mx = fmaxf(mx, y[tile]);
        }
        mx = row16_max(mx);
        float s0 = 0.f, s1 = 0.f;
        #pragma unroll
        for (int tile = 0; tile < 7; ++tile) {
            int t = tile * 16 + colL;
            if (t < T_STEPS) {
                float e = __expf(y[tile] - mx);
                s0 += e;
                s1 = fmaf(e, (float)t, s1);       // times = t * DT, DT = 1
            }
        }
        s0 = row16_sum(s0);
        s1 = row16_sum(s1);
        if (colL == 0)
            out[m0 + j + (hi ? 8 : 0)] = s1 / s0;
    }
}

extern "C" void kernel_launch(void* const* d_in, const int* in_sizes, int n_in,
                              void* d_out, int out_size, void* d_ws, size_t ws_size,
                              hipStream_t stream) {
    (void)in_sizes; (void)n_in; (void)out_size; (void)d_ws; (void)ws_size;
    const float* spk = (const float*)d_in[0];   // [512, 1024] f32
    const float* W   = (const float*)d_in[1];   // [256, 1024] f32
    float* out = (float*)d_out;                 // [512, 256]  f32

    const int rows = BATCH * N_OUT;             // 131072
    const int rows_per_block = 8 * 16;          // 8 waves x 16 rows
    dim3 grid(rows / rows_per_block);           // 1024 blocks
    dendritic_wmma_kernel<<<grid, 256, 0, stream>>>(spk, W, out);
}